// SoftRankingLoss_395136991775
// MI455X (gfx1250) — compile-verified
//
#include <hip/hip_runtime.h>

typedef float v2f __attribute__((ext_vector_type(2)));
typedef float v8f __attribute__((ext_vector_type(8)));

#define Bn 1024
#define Cn 3862
#define Nn (Bn * Cn)          // 3,954,688
#define PPR 5
#define Pn (Bn * PPR)         // 5120 positives
#define TOPNEG 30
#define Kn (Bn * TOPNEG)      // 30720 selected negatives

#define C_PREFIX 0
#define C_REMAIN 1
#define C_CGT    2
#define C_EQNEED 3

// order-preserving float -> uint key (ascending)
__device__ __forceinline__ unsigned fkey(float f) {
  unsigned u = __float_as_uint(f);
  return (u & 0x80000000u) ? ~u : (u | 0x80000000u);
}

// softplus(d + 1), numerically stable
__device__ __forceinline__ float softplus1(float d) {
  float x = d + 1.0f;
  float e = __expf(-fabsf(x));
  return fmaxf(x, 0.0f) + __logf(1.0f + e);
}

// ---------------- init: zero control + histogram ----------------
__global__ void k_init(unsigned* ctrl, unsigned* hist) {
  int t = threadIdx.x;                 // 256 threads
  if (t == 0) {
    for (int i = 0; i < 16; ++i) ctrl[i] = 0;
    ctrl[C_REMAIN] = Kn;
  }
  hist[t] = 0;
}

// ---------------- positives: stable per-row wave extraction ----------------
__global__ void k_pos(const float* __restrict__ pred, const int* __restrict__ tgt,
                      float* __restrict__ pos) {
  int row  = blockIdx.x;               // 1024 rows, 32 threads (1 wave)
  int lane = threadIdx.x;
  const float* pr = pred + row * Cn;
  const int*   tr = tgt  + row * Cn;
  int base = 0;
  unsigned lt = (1u << lane) - 1u;
  for (int c0 = 0; c0 < Cn; c0 += 32) {
    int col = c0 + lane;
    bool p = (col < Cn) && (tr[col] > 0);
    unsigned m = (unsigned)(__ballot(p) & 0xffffffffull);
    if (p) {
      int idx = row * PPR + base + __popc(m & lt);
      if (idx < Pn) pos[idx] = pr[col];
    }
    base += __popc(m);
  }
}

// ---------------- radix-select histogram (8 bits/round, MSB first) ----------------
__global__ void k_hist(const float* __restrict__ pred, const int* __restrict__ tgt,
                       const unsigned* __restrict__ ctrl, unsigned* __restrict__ hist,
                       int shift) {
  __shared__ unsigned lh[256];
  int t = threadIdx.x;
  lh[t] = 0;
  __syncthreads();
  unsigned prefix = ctrl[C_PREFIX];
  unsigned maskHi = (shift >= 24) ? 0u : (0xFFFFFFFFu << (shift + 8));
  int stride = gridDim.x * blockDim.x;
  for (int i = blockIdx.x * blockDim.x + t; i < Nn; i += stride) {
    if (tgt[i] > 0) continue;
    unsigned k = fkey(pred[i]);
    if ((k & maskHi) == (prefix & maskHi))
      atomicAdd(&lh[(k >> shift) & 255u], 1u);
  }
  __syncthreads();
  if (lh[t]) atomicAdd(&hist[t], lh[t]);
}

// ---------------- pick bin of K-th largest; reset histogram ----------------
__global__ void k_scan(unsigned* ctrl, unsigned* hist, int shift, int last) {
  int t = threadIdx.x;                 // 256 threads
  if (t == 0) {
    unsigned r = ctrl[C_REMAIN];
    unsigned cum = 0;
    unsigned prefix = ctrl[C_PREFIX];
    for (int b = 255; b >= 0; --b) {
      unsigned c = hist[b];
      if (cum + c >= r) { prefix |= ((unsigned)b) << shift; r -= cum; break; }
      cum += c;
    }
    ctrl[C_PREFIX] = prefix;
    ctrl[C_REMAIN] = r;
    if (last) { ctrl[C_CGT] = Kn - r; ctrl[C_EQNEED] = r; }
  }
  __syncthreads();
  hist[t] = 0;
}

// ---------------- per-block (=per-row) counts of >thr / ==thr ----------------
__global__ void k_count(const float* __restrict__ pred, const int* __restrict__ tgt,
                        const unsigned* __restrict__ ctrl,
                        unsigned* __restrict__ gtc, unsigned* __restrict__ eqc) {
  __shared__ unsigned sg[256], se[256];
  int b = blockIdx.x, t = threadIdx.x;
  unsigned thr = ctrl[C_PREFIX];
  int start = b * Cn;
  unsigned g = 0, e = 0;
  for (int i = t; i < Cn; i += 256) {
    int idx = start + i;
    if (tgt[idx] > 0) continue;
    unsigned k = fkey(pred[idx]);
    g += (k > thr);
    e += (k == thr);
  }
  sg[t] = g; se[t] = e;
  __syncthreads();
  for (int s = 128; s > 0; s >>= 1) {
    if (t < s) { sg[t] += sg[t + s]; se[t] += se[t + s]; }
    __syncthreads();
  }
  if (t == 0) { gtc[b] = sg[0]; eqc[b] = se[0]; }
}

// ---------------- exclusive scan of block counts (tiny, serial) ----------------
__global__ void k_offs(const unsigned* __restrict__ gtc, const unsigned* __restrict__ eqc,
                       unsigned* __restrict__ gto, unsigned* __restrict__ eqo) {
  if (threadIdx.x == 0) {
    unsigned cg = 0, ce = 0;
    for (int b = 0; b < Bn; ++b) {
      gto[b] = cg; cg += gtc[b];
      eqo[b] = ce; ce += eqc[b];
    }
  }
}

// ---------------- stable compaction of selected negatives ----------------
__global__ void k_compact(const float* __restrict__ pred, const int* __restrict__ tgt,
                          const unsigned* __restrict__ ctrl,
                          const unsigned* __restrict__ gto, const unsigned* __restrict__ eqo,
                          float* __restrict__ neg) {
  __shared__ unsigned wg[8], we[8];
  int b = blockIdx.x, t = threadIdx.x;   // 1024 blocks x 256 threads
  int wave = t >> 5, lane = t & 31;
  unsigned lt = (1u << lane) - 1u;
  unsigned thr = ctrl[C_PREFIX];
  unsigned cgt = ctrl[C_CGT];
  unsigned eqneed = ctrl[C_EQNEED];
  int start = b * Cn;
  unsigned gbase = gto[b], ebase = eqo[b];
  for (int i0 = 0; i0 < Cn; i0 += 256) {
    int i = i0 + t;
    float f = 0.f; bool isg = false, ise = false;
    if (i < Cn) {
      int idx = start + i;
      if (tgt[idx] <= 0) {
        f = pred[idx];
        unsigned k = fkey(f);
        isg = (k > thr);
        ise = (k == thr);
      }
    }
    unsigned mg = (unsigned)(__ballot(isg) & 0xffffffffull);
    unsigned me = (unsigned)(__ballot(ise) & 0xffffffffull);
    if (lane == 0) { wg[wave] = __popc(mg); we[wave] = __popc(me); }
    __syncthreads();
    unsigned pg = 0, pe = 0, tgsum = 0, tesum = 0;
    for (int w = 0; w < 8; ++w) {
      if (w < wave) { pg += wg[w]; pe += we[w]; }
      tgsum += wg[w]; tesum += we[w];
    }
    if (isg) neg[gbase + pg + __popc(mg & lt)] = f;
    if (ise) {
      unsigned r = ebase + pe + __popc(me & lt);
      if (r < eqneed) neg[cgt + r] = f;
    }
    gbase += tgsum; ebase += tesum;
    __syncthreads();
  }
}

// ---------------- pairwise softplus over pos x neg; WMMA wave reduction ----------------
#define POS_TILE 64
#define NEG_TILE 1024
#define NEG_TILES (Kn / NEG_TILE)      // 30
#define POS_TILES (Pn / POS_TILE)      // 80
#define NBLK (NEG_TILES * POS_TILES)   // 2400

__global__ void k_pairs(const float* __restrict__ pos, const float* __restrict__ neg,
                        float* __restrict__ psum, float* __restrict__ peq) {
  __shared__ float spv[POS_TILE];
  __shared__ float rs[16], re[16];
  int t = threadIdx.x;                 // 256 threads = 8 waves
  int nb = blockIdx.x % NEG_TILES;
  int pb = blockIdx.x / NEG_TILES;
  if (t < POS_TILE) spv[t] = pos[pb * POS_TILE + t];
  float nv0 = neg[nb * NEG_TILE + t];
  float nv1 = neg[nb * NEG_TILE + t + 256];
  float nv2 = neg[nb * NEG_TILE + t + 512];
  float nv3 = neg[nb * NEG_TILE + t + 768];
  __syncthreads();

  float acc = 0.f, eq = 0.f;
  #pragma unroll 4
  for (int j = 0; j < POS_TILE; ++j) {
    float p = spv[j];
    float d0 = nv0 - p, d1 = nv1 - p, d2 = nv2 - p, d3 = nv3 - p;
    acc += (d0 != 0.f) ? softplus1(d0) : 0.f;  eq += (d0 == 0.f) ? 1.f : 0.f;
    acc += (d1 != 0.f) ? softplus1(d1) : 0.f;  eq += (d1 == 0.f) ? 1.f : 0.f;
    acc += (d2 != 0.f) ? softplus1(d2) : 0.f;  eq += (d2 == 0.f) ? 1.f : 0.f;
    acc += (d3 != 0.f) ? softplus1(d3) : 0.f;  eq += (d3 == 0.f) ? 1.f : 0.f;
  }

  // Wave-level f32 reduction via V_WMMA_F32_16X16X4_F32 with B = ones:
  // A: lane L holds A[L][0] (vgpr0) / A[L][1] (vgpr1); lane L+16 holds A[L][2]/A[L][3].
  // With a[1] = 0 and B all-ones: D[m][n] = acc_m + acc_{m+16} (exact f32, x1.0 multiply).
  v2f a, bmat;
  v8f cz = {};
  bmat[0] = 1.0f; bmat[1] = 1.0f;
  a[0] = acc; a[1] = 0.0f;
  v8f dS = __builtin_amdgcn_wmma_f32_16x16x4_f32(false, a, false, bmat, (short)0, cz, false, false);
  a[0] = eq;  a[1] = 0.0f;
  v8f dE = __builtin_amdgcn_wmma_f32_16x16x4_f32(false, a, false, bmat, (short)0, cz, false, false);
  float s1 = dS[0] + dS[1] + dS[2] + dS[3] + dS[4] + dS[5] + dS[6] + dS[7];
  float s2 = dE[0] + dE[1] + dE[2] + dE[3] + dE[4] + dE[5] + dE[6] + dE[7];

  int wave = t >> 5, lane = t & 31;
  if (lane == 0 || lane == 16) {       // column 0 rows 0..7 (lane 0) and 8..15 (lane 16)
    rs[wave * 2 + (lane >> 4)] = s1;
    re[wave * 2 + (lane >> 4)] = s2;
  }
  __syncthreads();
  if (t == 0) {
    float S = 0.f, E = 0.f;
    for (int i = 0; i < 16; ++i) { S += rs[i]; E += re[i]; }
    psum[blockIdx.x] = S;
    peq[blockIdx.x] = E;
  }
}

// ---------------- deterministic final reduction ----------------
__global__ void k_final(const float* __restrict__ psum, const float* __restrict__ peq,
                        float* __restrict__ out) {
  __shared__ float ss[256], se[256];
  int t = threadIdx.x;
  float S = 0.f, E = 0.f;
  for (int i = t; i < NBLK; i += 256) { S += psum[i]; E += peq[i]; }
  ss[t] = S; se[t] = E;
  __syncthreads();
  if (t == 0) {
    float tot = 0.f, eqt = 0.f;
    for (int i = 0; i < 256; ++i) { tot += ss[i]; eqt += se[i]; }
    long long cnt = (long long)Pn * (long long)Kn - (long long)(eqt + 0.5f);
    out[0] = tot / (float)cnt;
  }
}

extern "C" void kernel_launch(void* const* d_in, const int* in_sizes, int n_in,
                              void* d_out, int out_size, void* d_ws, size_t ws_size,
                              hipStream_t stream) {
  const float* pred = (const float*)d_in[0];
  const int*   tgt  = (const int*)d_in[1];
  // d_in[2] = top_neg_count (30), d_in[3] = pos_count (5120): static per the reference.
  float* out = (float*)d_out;

  unsigned* ws   = (unsigned*)d_ws;
  unsigned* ctrl = ws;                  // 16
  unsigned* hist = ws + 16;             // 256
  unsigned* gtc  = ws + 272;            // 1024
  unsigned* eqc  = ws + 1296;           // 1024
  unsigned* gto  = ws + 2320;           // 1024
  unsigned* eqo  = ws + 3344;           // 1024
  float* pos  = (float*)(ws + 4368);    // 5120
  float* neg  = (float*)(ws + 9488);    // 30720
  float* psum = (float*)(ws + 40208);   // 2400
  float* peq  = (float*)(ws + 42608);   // 2400  (end ~176 KB)

  k_init<<<1, 256, 0, stream>>>(ctrl, hist);
  k_pos<<<Bn, 32, 0, stream>>>(pred, tgt, pos);
  for (int r = 0; r < 4; ++r) {
    int shift = 24 - 8 * r;
    k_hist<<<512, 256, 0, stream>>>(pred, tgt, ctrl, hist, shift);
    k_scan<<<1, 256, 0, stream>>>(ctrl, hist, shift, r == 3);
  }
  k_count<<<Bn, 256, 0, stream>>>(pred, tgt, ctrl, gtc, eqc);
  k_offs<<<1, 64, 0, stream>>>(gtc, eqc, gto, eqo);
  k_compact<<<Bn, 256, 0, stream>>>(pred, tgt, ctrl, gto, eqo, neg);
  k_pairs<<<NBLK, 256, 0, stream>>>(pos, neg, psum, peq);
  k_final<<<1, 256, 0, stream>>>(psum, peq, out);
}